// learnable_skinning_layer_25769804116
// MI455X (gfx1250) — compile-verified
//
#include <hip/hip_runtime.h>
#include <math.h>

// Problem constants (match reference)
#define VNUM     100000
#define KNZ      8               // nonzero bases per vertex
#define DIM      9
#define BATCH    16
#define BASE_NUM 64
#define NFEAT    (BATCH * DIM)   // 144 output features per vertex
#define WAVES    4               // waves per block (wave32)
#define VPW      16              // vertices per wave (one WMMA M-tile)
#define VPB      (WAVES * VPW)   // 64 vertices per block

typedef float v2f __attribute__((ext_vector_type(2)));
typedef float v8f __attribute__((ext_vector_type(8)));

// gfx1250 async global<->LDS copy engine (ASYNCcnt-tracked), with fallback.
#if __has_builtin(__builtin_amdgcn_global_load_async_to_lds_b32) && \
    __has_builtin(__builtin_amdgcn_global_store_async_from_lds_b32)
#define HAVE_ASYNC_LDS 1
#else
#define HAVE_ASYNC_LDS 0
#endif

// Builtins take address-space-qualified int* (per hipcc diagnostic).
typedef __attribute__((address_space(1))) int as1_int;   // global
typedef __attribute__((address_space(3))) int as3_int;   // LDS

__device__ __forceinline__ void wait_asynccnt0() {
    asm volatile("s_wait_asynccnt 0x0" ::: "memory");
}
__device__ __forceinline__ void wait_dscnt0() {
    asm volatile("s_wait_dscnt 0x0" ::: "memory");
}

// Out[b, v, d] = sum_base W[v, base] * F[base, b*9+d]
// W is the densified segment-softmax weight matrix (8 nonzeros -> 64 slots).
__global__ __launch_bounds__(WAVES * 32)
void skin_wmma_kernel(const float* __restrict__ base_fs,   // [16, 64*9]
                      const float* __restrict__ ws,        // [VNUM*8]
                      const int*   __restrict__ vb_index,  // [VNUM*8], 0..63
                      float*       __restrict__ out)       // [16, VNUM, 9]
{
    // F staged as [base][feature], feature = b*DIM + d      (64 x 144 f32 = 36 KB)
    __shared__ float F[BASE_NUM][NFEAT];
    // Dense per-vertex weight rows, one 16x64 tile per wave  (4 x 16 x 64 f32 = 16 KB)
    __shared__ float W[WAVES][VPW][BASE_NUM];
    // D-tile transpose buffer in [vertex][feature] order     (4 x 16 x 144 f32 = 36 KB)
    __shared__ float T[WAVES][VPW][NFEAT];

    const int tid  = threadIdx.x;      // 0..127
    const int wave = tid >> 5;
    const int lane = tid & 31;
    const int vblk = blockIdx.x * VPB; // first vertex of this block

    // ---- zero the dense weight tiles ----
    for (int i = tid; i < WAVES * VPW * BASE_NUM; i += blockDim.x)
        ((float*)W)[i] = 0.0f;

    // ---- stage base_fs into LDS (transposed to [base][b*9+d]) ----
    for (int i = tid; i < BASE_NUM * NFEAT; i += blockDim.x) {
        int base = i / NFEAT;
        int f    = i - base * NFEAT;
        int b    = f / DIM;
        int d    = f - b * DIM;
        const float* src = base_fs + b * (BASE_NUM * DIM) + base * DIM + d;
#if HAVE_ASYNC_LDS
        __builtin_amdgcn_global_load_async_to_lds_b32(
            (as1_int*)src, (as3_int*)&F[base][f], 0, 0);
#else
        F[base][f] = *src;
#endif
    }
#if HAVE_ASYNC_LDS
    wait_asynccnt0();                  // each wave drains its own ASYNCcnt
#endif
    __syncthreads();

    // ---- fused segment softmax (8 contiguous ws per vertex) + scatter to dense W ----
    if (tid < VPB) {
        int v = vblk + tid;
        if (v < VNUM) {
            const float* wp = ws + (long)v * KNZ;
            float e[KNZ];
            float m = -INFINITY;
            #pragma unroll
            for (int j = 0; j < KNZ; ++j) { e[j] = wp[j]; m = fmaxf(m, e[j]); }
            float s = 0.0f;
            #pragma unroll
            for (int j = 0; j < KNZ; ++j) { e[j] = __expf(e[j] - m); s += e[j]; }
            float inv = 1.0f / s;
            const int* bp = vb_index + (long)v * KNZ;
            int wv = tid >> 4, row = tid & 15;
            #pragma unroll
            for (int j = 0; j < KNZ; ++j)
                atomicAdd(&W[wv][row][bp[j]], e[j] * inv);  // ds_add_f32; handles dup bases
        }
    }
    __syncthreads();

    // ---- WMMA: each wave computes its 16 vertices x 144 features ----
    // A-matrix 16x4 f32 (ISA 7.12.2): lanes 0-15 M=lane (K 0,1); lanes 16-31 M=lane-16 (K 2,3)
    const int m      = lane & 15;
    const int khalf  = (lane < 16) ? 0 : 2;
    const int rowoff = (lane < 16) ? 0 : 8;   // C/D: VGPR r -> M=r (lo half) / M=r+8 (hi half)
    const float* Wrow = &W[wave][m][0];
    const int vwave  = vblk + wave * VPW;

    for (int t = 0; t < NFEAT / 16; ++t) {    // 9 feature tiles of 16 columns
        const int fcol = t * 16 + m;          // this lane's N column
        v8f c = {};
        #pragma unroll
        for (int k = 0; k < BASE_NUM / 4; ++k) {          // 16 K-steps of depth 4
            v2f a = *(const v2f*)(Wrow + 4 * k + khalf);  // K = khalf, khalf+1
            v2f b;
            b.x = F[4 * k + khalf + 0][fcol];             // B rows striped like C/D
            b.y = F[4 * k + khalf + 1][fcol];
            c = __builtin_amdgcn_wmma_f32_16x16x4_f32(
                    /*neg_a=*/false, a, /*neg_b=*/false, b,
                    /*c_mod=*/(short)0, c, /*reuse_a=*/false, /*reuse_b=*/false);
        }
        // Transpose D tile into LDS [vertex][feature] order (ds_store_b32 x8)
        #pragma unroll
        for (int r = 0; r < 8; ++r)
            T[wave][rowoff + r][fcol] = c[r];
    }

    // DS ops and ASYNC ops are not mutually ordered: drain DScnt before the
    // async store engine reads T. (Also acts as a compiler memory barrier.)
    wait_dscnt0();

    // ---- coalesced output: per batch, 144 contiguous floats per wave ----
    for (int b = 0; b < BATCH; ++b) {
        float* dst = out + (long)b * (VNUM * DIM) + (long)vwave * DIM;
        for (int j = lane; j < VPW * DIM; j += 32) {      // 144 floats, lane-consecutive
            int v = j / DIM;
            int d = j - v * DIM;
            if (vwave + v < VNUM) {
#if HAVE_ASYNC_LDS
                __builtin_amdgcn_global_store_async_from_lds_b32(
                    (as1_int*)(dst + j), (as3_int*)&T[wave][v][b * DIM + d], 0, 0);
#else
                dst[j] = T[wave][v][b * DIM + d];
#endif
            }
        }
    }
    // s_endpgm performs an implicit wait-idle, covering outstanding async stores.
}

extern "C" void kernel_launch(void* const* d_in, const int* in_sizes, int n_in,
                              void* d_out, int out_size, void* d_ws, size_t ws_size,
                              hipStream_t stream) {
    const float* base_fs  = (const float*)d_in[0];
    const float* ws       = (const float*)d_in[1];
    const int*   vb_index = (const int*)d_in[2];
    // d_in[3] (vv_index) is structurally repeat(arange(VNUM), 8); not needed.
    float* out = (float*)d_out;

    dim3 grid((VNUM + VPB - 1) / VPB);   // 1563 blocks
    dim3 block(WAVES * 32);              // 128 threads = 4 wave32
    hipLaunchKernelGGL(skin_wmma_kernel, grid, block, 0, stream,
                       base_fs, ws, vb_index, out);
}